// MTAD_MODEL3_85607288143887
// MI455X (gfx1250) — compile-verified
//
#include <hip/hip_runtime.h>
#include <hip/hip_bf16.h>
#include <math.h>

// Problem constants (match reference)
#define BB    64
#define WW    128
#define NN    64
#define EGLC  12
#define HIDC  77
#define ETC   128
#define BnC   (BB * NN)      // 4096
#define KCATC (WW + EGLC)    // 140
#define SLOPEC 0.2f
#define OUTREC_ELEMS ((size_t)BB * WW * NN)  // 524288

typedef float v2f __attribute__((ext_vector_type(2)));
typedef float v8f __attribute__((ext_vector_type(8)));

// ---------------------------------------------------------------------------
// WMMA helper: D = A(16x4 f32) x B(4x16 f32) + C, full-precision f32 path.
// Per ISA layout: A frag lane L (l=L&15, hi=L>>4): v0=A[l][k+2hi], v1=A[l][k+2hi+1]
// B frag: v0=B[k+2hi][l], v1=B[k+2hi+1][l]. C/D: vgpr r -> row r+8*hi, col l.
// ---------------------------------------------------------------------------
__device__ __forceinline__ v8f wmma4(v2f a, v2f b, v8f c) {
  return __builtin_amdgcn_wmma_f32_16x16x4_f32(false, a, false, b, (short)0, c,
                                               false, false);
}

// ---------------------------------------------------------------------------
// Utility kernels
// ---------------------------------------------------------------------------
__global__ void k_fill(float* p, int n, float v) {
  int i = blockIdx.x * blockDim.x + threadIdx.x;
  if (i < n) p[i] = v;
}

__global__ void k_fill_u32(unsigned* p, int n, unsigned v) {
  int i = blockIdx.x * blockDim.x + threadIdx.x;
  if (i < n) p[i] = v;
}

// Build xcat (Bn x 140): [x_batch | tiled gl_embed]
// x is (B, W, N); x_batch[b*N+n][w] = x[b][w][n]
__global__ void k_prep(const float* __restrict__ x, const float* __restrict__ gl,
                       float* __restrict__ xcat) {
  int i = blockIdx.x * blockDim.x + threadIdx.x;
  if (i >= BnC * KCATC) return;
  int row = i / KCATC, c = i % KCATC;
  int b = row >> 6, n = row & 63;
  float v = (c < WW) ? x[((size_t)b * WW + c) * NN + n]
                     : gl[n * EGLC + (c - WW)];
  xcat[i] = v;
}

// GCN degree: deg[dst] += 1 per edge (deg pre-filled with 1.0 for self loop)
__global__ void k_deg(const int* __restrict__ ei, int nE, float* deg) {
  int e = blockIdx.x * blockDim.x + threadIdx.x;
  if (e < nE) atomicAdd(&deg[ei[nE + e]], 1.0f);
}

__global__ void k_dinv(float* deg) {
  int i = blockIdx.x * blockDim.x + threadIdx.x;
  if (i < BnC) deg[i] = rsqrtf(fmaxf(deg[i], 1.0f));
}

// ---------------------------------------------------------------------------
// Standard GEMM: C(Bn x 128) = A(Bn x K, lda) * B(K x 128). One wave per tile.
// 256 x 8 tiles -> 2048 waves -> <<<512,128>>>.
// ---------------------------------------------------------------------------
__global__ void k_gemm_std(const float* __restrict__ A0, int lda, int K,
                           const float* __restrict__ Bm, float* __restrict__ C) {
  int gw = (blockIdx.x * blockDim.x + threadIdx.x) >> 5;
  int lane = threadIdx.x & 31;
  int tm = gw >> 3, tn = gw & 7;
  int l = lane & 15, hi = lane >> 4;
  const float* A = A0 + (size_t)tm * 16 * lda;
  const float* Bp = Bm + tn * 16;
  v8f acc0 = {}, acc1 = {};
  int k = 0;
  for (; k + 8 <= K; k += 8) {
    v2f a, b;
    a.x = A[l * lda + k + 2 * hi];       a.y = A[l * lda + k + 2 * hi + 1];
    b.x = Bp[(k + 2 * hi) * WW + l];     b.y = Bp[(k + 2 * hi + 1) * WW + l];
    acc0 = wmma4(a, b, acc0);
    a.x = A[l * lda + k + 4 + 2 * hi];   a.y = A[l * lda + k + 4 + 2 * hi + 1];
    b.x = Bp[(k + 4 + 2 * hi) * WW + l]; b.y = Bp[(k + 4 + 2 * hi + 1) * WW + l];
    acc1 = wmma4(a, b, acc1);
  }
  for (; k < K; k += 4) {
    v2f a, b;
    a.x = A[l * lda + k + 2 * hi];   a.y = A[l * lda + k + 2 * hi + 1];
    b.x = Bp[(k + 2 * hi) * WW + l]; b.y = Bp[(k + 2 * hi + 1) * WW + l];
    acc0 = wmma4(a, b, acc0);
  }
  v8f acc = acc0 + acc1;
  float* Cp = C + (size_t)tm * 16 * WW + tn * 16;
  for (int r = 0; r < 8; r++) Cp[(r + 8 * hi) * WW + l] = acc[r];
}

// GCN aggregation: out[dst] += h[src] * dinv[src]*dinv[dst] (edges + self loops)
__global__ void k_gcn_agg(const int* __restrict__ ei, int nE,
                          const float* __restrict__ h,
                          const float* __restrict__ dinv, float* out) {
  int i = blockIdx.x * blockDim.x + threadIdx.x;
  int e = i >> 7, c = i & 127;
  int total = nE + BnC;
  if (e >= total) return;
  int s, d;
  if (e < nE) { s = ei[e]; d = ei[nE + e]; } else { s = d = e - nE; }
  float coef = dinv[s] * dinv[d];
  atomicAdd(&out[(size_t)d * WW + c], h[(size_t)s * WW + c] * coef);
}

__global__ void k_bias_relu(float* h, const float* __restrict__ b) {
  int i = blockIdx.x * blockDim.x + threadIdx.x;
  if (i >= BnC * WW) return;
  h[i] = fmaxf(h[i] + b[i & 127], 0.0f);
}

// MLP layer 1: a1(Bn x 80, cols 77..79 = 0) = tanh(h(Bn x 128) @ w1(128 x 77) + b1)
// 256 x 5 tiles -> 1280 waves -> <<<320,128>>>
__global__ void k_gemm_a1(const float* __restrict__ h, const float* __restrict__ w1,
                          const float* __restrict__ b1, float* __restrict__ a1) {
  int gw = (blockIdx.x * blockDim.x + threadIdx.x) >> 5;
  int lane = threadIdx.x & 31;
  int tm = gw / 5, tn = gw % 5;
  int l = lane & 15, hi = lane >> 4;
  int col = tn * 16 + l;
  const float* A = h + (size_t)tm * 16 * WW;
  v8f acc0 = {}, acc1 = {};
  for (int k = 0; k < WW; k += 8) {
    v2f a, b;
    a.x = A[l * WW + k + 2 * hi];     a.y = A[l * WW + k + 2 * hi + 1];
    b.x = (col < HIDC) ? w1[(k + 2 * hi) * HIDC + col] : 0.0f;
    b.y = (col < HIDC) ? w1[(k + 2 * hi + 1) * HIDC + col] : 0.0f;
    acc0 = wmma4(a, b, acc0);
    a.x = A[l * WW + k + 4 + 2 * hi]; a.y = A[l * WW + k + 4 + 2 * hi + 1];
    b.x = (col < HIDC) ? w1[(k + 4 + 2 * hi) * HIDC + col] : 0.0f;
    b.y = (col < HIDC) ? w1[(k + 4 + 2 * hi + 1) * HIDC + col] : 0.0f;
    acc1 = wmma4(a, b, acc1);
  }
  v8f acc = acc0 + acc1;
  float bias = (col < HIDC) ? b1[col] : 0.0f;
  float* C = a1 + (size_t)tm * 16 * 80 + tn * 16;
  for (int r = 0; r < 8; r++) {
    float v = (col < HIDC) ? tanhf(acc[r] + bias) : 0.0f;
    C[(r + 8 * hi) * 80 + l] = v;
  }
}

// Zero-pad mlp_w2 (77 x 128) into w2p (80 x 128)
__global__ void k_padw2(const float* __restrict__ w2, float* __restrict__ w2p) {
  int i = blockIdx.x * blockDim.x + threadIdx.x;
  if (i >= 80 * WW) return;
  w2p[i] = ((i >> 7) < HIDC) ? w2[i] : 0.0f;
}

// MLP layer 2 fused epilogue: x_rec = tanh(a1 @ w2p + b2);
// writes xs = x_batch + x_rec and out_rec[b][w][n] = x_rec[row][w]
__global__ void k_gemm_xrec(const float* __restrict__ a1, const float* __restrict__ w2p,
                            const float* __restrict__ b2, const float* __restrict__ xcat,
                            float* __restrict__ xs, float* __restrict__ outrec) {
  int gw = (blockIdx.x * blockDim.x + threadIdx.x) >> 5;
  int lane = threadIdx.x & 31;
  int tm = gw >> 3, tn = gw & 7;
  int l = lane & 15, hi = lane >> 4;
  const float* A = a1 + (size_t)tm * 16 * 80;
  const float* Bp = w2p + tn * 16;
  v8f acc0 = {}, acc1 = {};
  for (int k = 0; k < 80; k += 8) {
    v2f a, b;
    a.x = A[l * 80 + k + 2 * hi];        a.y = A[l * 80 + k + 2 * hi + 1];
    b.x = Bp[(k + 2 * hi) * WW + l];     b.y = Bp[(k + 2 * hi + 1) * WW + l];
    acc0 = wmma4(a, b, acc0);
    a.x = A[l * 80 + k + 4 + 2 * hi];    a.y = A[l * 80 + k + 4 + 2 * hi + 1];
    b.x = Bp[(k + 4 + 2 * hi) * WW + l]; b.y = Bp[(k + 4 + 2 * hi + 1) * WW + l];
    acc1 = wmma4(a, b, acc1);
  }
  v8f acc = acc0 + acc1;
  int col = tn * 16 + l;
  float bias = b2[col];
  for (int r = 0; r < 8; r++) {
    int row = tm * 16 + r + 8 * hi;                      // = b*N + n
    float v = tanhf(acc[r] + bias);
    xs[(size_t)row * WW + col] = xcat[(size_t)row * KCATC + col] + v;
    outrec[((size_t)(row >> 6) * WW + col) * NN + (row & 63)] = v;
  }
}

// Temporal attention projections: left = v@wi, right = v@wj, per batch.
// v[b][m][k] = xs[(b*N+k)*W + m] (transposed access). M=128, K=64, N=128.
// 64 batches x 8x8 tiles -> 4096 waves -> <<<1024,128>>>. A-frag reused for both.
__global__ void k_gemm_lr(const float* __restrict__ xs, const float* __restrict__ taw,
                          float* __restrict__ left, float* __restrict__ right) {
  int gw = (blockIdx.x * blockDim.x + threadIdx.x) >> 5;
  int lane = threadIdx.x & 31;
  int b = gw >> 6, t = gw & 63, tm = t >> 3, tn = t & 7;
  int l = lane & 15, hi = lane >> 4;
  const float* A  = xs + (size_t)b * NN * WW + tm * 16;   // (m,k) at A[k*W + m]
  const float* B1 = taw + tn * 16;                        // wi rows 0..63
  const float* B2 = taw + (size_t)NN * ETC + tn * 16;     // wj rows 64..127
  v8f accL = {}, accR = {};
  for (int k = 0; k < NN; k += 4) {
    v2f a, b1, b2;
    a.x  = A[(k + 2 * hi) * WW + l];      a.y  = A[(k + 2 * hi + 1) * WW + l];
    b1.x = B1[(k + 2 * hi) * ETC + l];    b1.y = B1[(k + 2 * hi + 1) * ETC + l];
    b2.x = B2[(k + 2 * hi) * ETC + l];    b2.y = B2[(k + 2 * hi + 1) * ETC + l];
    accL = wmma4(a, b1, accL);
    accR = wmma4(a, b2, accR);
  }
  float* L = left  + ((size_t)b * WW + tm * 16) * ETC + tn * 16;
  float* R = right + ((size_t)b * WW + tm * 16) * ETC + tn * 16;
  for (int r = 0; r < 8; r++) {
    L[(r + 8 * hi) * ETC + l] = accL[r];
    R[(r + 8 * hi) * ETC + l] = accR[r];
  }
}

// e[b,i,j] = sum_e lrelu(L[b,i,e]+R[b,j,e]+tb[e])*ta[e] + bias[i,j]; softmax over j.
// One block per (b,i), 128 threads (thread = j). Fused softmax.
__global__ void k_attn(const float* __restrict__ left, const float* __restrict__ right,
                       const float* __restrict__ tb, const float* __restrict__ ta,
                       const float* __restrict__ bias, float* __restrict__ attn) {
  int b = blockIdx.x >> 7;       // /W
  int i = blockIdx.x & 127;
  int j = threadIdx.x;
  __shared__ float L[ETC], TB[ETC], TA[ETC], red[128];
  L[j]  = left[((size_t)b * WW + i) * ETC + j];
  TB[j] = tb[j];
  TA[j] = ta[j];
  __syncthreads();
  const float* R = right + ((size_t)b * WW + j) * ETC;
  float e = 0.0f;
  for (int k = 0; k < ETC; k++) {
    float s = L[k] + R[k] + TB[k];
    s = (s > 0.0f) ? s : SLOPEC * s;
    e += s * TA[k];
  }
  e += bias[i * WW + j];
  red[j] = e; __syncthreads();
  for (int st = 64; st > 0; st >>= 1) {
    if (j < st) red[j] = fmaxf(red[j], red[j + st]);
    __syncthreads();
  }
  float m = red[0]; __syncthreads();
  float ex = expf(e - m);
  red[j] = ex; __syncthreads();
  for (int st = 64; st > 0; st >>= 1) {
    if (j < st) red[j] += red[j + st];
    __syncthreads();
  }
  attn[((size_t)b * WW + i) * WW + j] = ex / red[0];
}

// ht = tanh(attn @ v) stored transposed as hg[(b*N+n)*W + i].
// Per batch: M=128(i), K=128(j), N=64(n); B(k=j,n) = xs[(b*N+n)*W + j].
// 64 x (8x4) tiles -> 2048 waves -> <<<512,128>>>
__global__ void k_gemm_av(const float* __restrict__ attn, const float* __restrict__ xs,
                          float* __restrict__ hg) {
  int gw = (blockIdx.x * blockDim.x + threadIdx.x) >> 5;
  int lane = threadIdx.x & 31;
  int b = gw >> 5, t = gw & 31, tm = t >> 2, tn = t & 3;
  int l = lane & 15, hi = lane >> 4;
  const float* A  = attn + (size_t)b * WW * WW + (size_t)tm * 16 * WW;
  const float* Bp = xs + (size_t)b * NN * WW + (size_t)tn * 16 * WW; // (k,n) at Bp[n*W + k]
  v8f acc0 = {}, acc1 = {};
  for (int k = 0; k < WW; k += 8) {
    v2f a, bv;
    a.x  = A[l * WW + k + 2 * hi];      a.y  = A[l * WW + k + 2 * hi + 1];
    bv.x = Bp[l * WW + k + 2 * hi];     bv.y = Bp[l * WW + k + 2 * hi + 1];
    acc0 = wmma4(a, bv, acc0);
    a.x  = A[l * WW + k + 4 + 2 * hi];  a.y  = A[l * WW + k + 4 + 2 * hi + 1];
    bv.x = Bp[l * WW + k + 4 + 2 * hi]; bv.y = Bp[l * WW + k + 4 + 2 * hi + 1];
    acc1 = wmma4(a, bv, acc1);
  }
  v8f acc = acc0 + acc1;
  float* C = hg + ((size_t)b * NN + tn * 16) * WW + tm * 16; // [n_local*W + i_local]
  for (int r = 0; r < 8; r++) C[(size_t)l * WW + r + 8 * hi] = tanhf(acc[r]);
}

// GAT per-row scalars: ssrc[i] = h[i]@asrc, sdst[i] = h[i]@adst. One wave per row.
__global__ void k_srcdst(const float* __restrict__ h, const float* __restrict__ asrc,
                         const float* __restrict__ adst, float* ssrc, float* sdst) {
  int row = (blockIdx.x * blockDim.x + threadIdx.x) >> 5;
  int lane = threadIdx.x & 31;
  if (row >= BnC) return;
  float s1 = 0.0f, s2 = 0.0f;
  for (int k = lane; k < WW; k += 32) {
    float v = h[(size_t)row * WW + k];
    s1 += v * asrc[k];
    s2 += v * adst[k];
  }
  for (int off = 16; off > 0; off >>= 1) {
    s1 += __shfl_xor(s1, off, 32);
    s2 += __shfl_xor(s2, off, 32);
  }
  if (lane == 0) { ssrc[row] = s1; sdst[row] = s2; }
}

// Monotonic float<->uint mapping for atomic max on floats
__device__ __forceinline__ unsigned mapf(float f) {
  unsigned u = __float_as_uint(f);
  return u ^ ((((int)u) >> 31) | 0x80000000u);
}
__device__ __forceinline__ float unmapf(unsigned m) {
  return (m & 0x80000000u) ? __uint_as_float(m ^ 0x80000000u)
                           : __uint_as_float(~m);
}

// Edge logits + segment max (edges + self loops)
__global__ void k_edge_e(const int* __restrict__ ei, int nE,
                         const float* __restrict__ ssrc, const float* __restrict__ sdst,
                         float* __restrict__ eedge, unsigned* __restrict__ emax) {
  int e = blockIdx.x * blockDim.x + threadIdx.x;
  int total = nE + BnC;
  if (e >= total) return;
  int s, d;
  if (e < nE) { s = ei[e]; d = ei[nE + e]; } else { s = d = e - nE; }
  float v = ssrc[s] + sdst[d];
  v = (v > 0.0f) ? v : SLOPEC * v;
  eedge[e] = v;
  atomicMax(&emax[d], mapf(v));
}

// ee = exp(e - emax[dst]); denom[dst] += ee
__global__ void k_edge_exp(const int* __restrict__ ei, int nE,
                           float* __restrict__ eedge, const unsigned* __restrict__ emax,
                           float* __restrict__ denom) {
  int e = blockIdx.x * blockDim.x + threadIdx.x;
  int total = nE + BnC;
  if (e >= total) return;
  int d = (e < nE) ? ei[nE + e] : (e - nE);
  float ee = expf(eedge[e] - unmapf(emax[d]));
  eedge[e] = ee;
  atomicAdd(&denom[d], ee);
}

// GAT aggregation: out[dst] += h[src] * (ee / denom[dst])
__global__ void k_gat_agg(const int* __restrict__ ei, int nE,
                          const float* __restrict__ eedge, const float* __restrict__ denom,
                          const float* __restrict__ h, float* __restrict__ out) {
  int i = blockIdx.x * blockDim.x + threadIdx.x;
  int e = i >> 7, c = i & 127;
  int total = nE + BnC;
  if (e >= total) return;
  int s, d;
  if (e < nE) { s = ei[e]; d = ei[nE + e]; } else { s = d = e - nE; }
  float alpha = eedge[e] / denom[d];
  atomicAdd(&out[(size_t)d * WW + c], h[(size_t)s * WW + c] * alpha);
}

// Final FC: x_pre[row] = tanh((gout[row] + gat_b) . fc_w + fc_b). One wave per row.
__global__ void k_fc(const float* __restrict__ gout, const float* __restrict__ gat_b,
                     const float* __restrict__ fc_w, const float* __restrict__ fc_b,
                     float* __restrict__ out) {
  int row = (blockIdx.x * blockDim.x + threadIdx.x) >> 5;
  int lane = threadIdx.x & 31;
  if (row >= BnC) return;
  float s = 0.0f;
  for (int k = lane; k < WW; k += 32)
    s += (gout[(size_t)row * WW + k] + gat_b[k]) * fc_w[k];
  for (int off = 16; off > 0; off >>= 1) s += __shfl_xor(s, off, 32);
  if (lane == 0) out[row] = tanhf(s + fc_b[0]);
}

// ---------------------------------------------------------------------------
extern "C" void kernel_launch(void* const* d_in, const int* in_sizes, int n_in,
                              void* d_out, int out_size, void* d_ws, size_t ws_size,
                              hipStream_t stream) {
  (void)n_in; (void)out_size; (void)ws_size;
  const float* x        = (const float*)d_in[0];
  const float* gl_emb   = (const float*)d_in[7];
  const float* gcn_w    = (const float*)d_in[8];
  const float* gcn_b    = (const float*)d_in[9];
  const float* mlp_w1   = (const float*)d_in[10];
  const float* mlp_b1   = (const float*)d_in[11];
  const float* mlp_w2   = (const float*)d_in[12];
  const float* mlp_b2   = (const float*)d_in[13];
  const float* ta_w     = (const float*)d_in[14];
  const float* ta_b     = (const float*)d_in[15];
  const float* ta_a     = (const float*)d_in[16];
  const float* ta_bias  = (const float*)d_in[17];
  const float* gat_w    = (const float*)d_in[18];
  const float* gat_asrc = (const float*)d_in[19];
  const float* gat_adst = (const float*)d_in[20];
  const float* gat_b    = (const float*)d_in[21];
  const float* fc_w     = (const float*)d_in[22];
  const float* fc_b     = (const float*)d_in[23];
  const int* ei_gat     = (const int*)d_in[24];
  const int* ei_gl      = (const int*)d_in[25];
  const int nE_gat = in_sizes[24] / 2;
  const int nE_gl  = in_sizes[25] / 2;

  float* ws = (float*)d_ws;
  size_t o = 0;
  float* xcat  = ws + o; o += (size_t)BnC * KCATC;
  float* hlin  = ws + o; o += (size_t)BnC * WW;
  float* hagg  = ws + o; o += (size_t)BnC * WW;
  float* a1b   = ws + o; o += (size_t)BnC * 80;
  float* w2p   = ws + o; o += (size_t)80 * WW;
  float* xsb   = ws + o; o += (size_t)BnC * WW;
  float* leftb = ws + o; o += (size_t)BB * WW * ETC;
  float* rightb= ws + o; o += (size_t)BB * WW * ETC;
  float* attnb = ws + o; o += (size_t)BB * WW * WW;
  float* hgb   = ws + o; o += (size_t)BnC * WW;
  float* hgatb = ws + o; o += (size_t)BnC * WW;
  float* goutb = ws + o; o += (size_t)BnC * WW;
  float* dinvb = ws + o; o += BnC;
  float* ssrcb = ws + o; o += BnC;
  float* sdstb = ws + o; o += BnC;
  unsigned* emaxb = (unsigned*)(ws + o); o += BnC;
  float* denomb = ws + o; o += BnC;
  float* eedgeb = ws + o; o += (size_t)(nE_gat + BnC);

  float* outrec = (float*)d_out;
  float* outpre = (float*)d_out + OUTREC_ELEMS;

  auto cdiv = [](int a, int b) { return (a + b - 1) / b; };

  // --- Stage 1: GCN over global-learned graph ---
  k_fill<<<cdiv(BnC, 256), 256, 0, stream>>>(dinvb, BnC, 1.0f);   // self-loop deg
  k_prep<<<cdiv(BnC * KCATC, 256), 256, 0, stream>>>(x, gl_emb, xcat);
  k_deg<<<cdiv(nE_gl, 256), 256, 0, stream>>>(ei_gl, nE_gl, dinvb);
  k_dinv<<<cdiv(BnC, 256), 256, 0, stream>>>(dinvb);
  k_gemm_std<<<512, 128, 0, stream>>>(xcat, KCATC, KCATC, gcn_w, hlin);
  k_fill<<<cdiv(BnC * WW, 256), 256, 0, stream>>>(hagg, BnC * WW, 0.0f);
  k_gcn_agg<<<cdiv((nE_gl + BnC) * WW, 256), 256, 0, stream>>>(ei_gl, nE_gl, hlin,
                                                               dinvb, hagg);
  k_bias_relu<<<cdiv(BnC * WW, 256), 256, 0, stream>>>(hagg, gcn_b);

  // --- Stage 2: MLP reconstruction (+ out_rec and xs in epilogue) ---
  k_gemm_a1<<<320, 128, 0, stream>>>(hagg, mlp_w1, mlp_b1, a1b);
  k_padw2<<<cdiv(80 * WW, 256), 256, 0, stream>>>(mlp_w2, w2p);
  k_gemm_xrec<<<512, 128, 0, stream>>>(a1b, w2p, mlp_b2, xcat, xsb, outrec);

  // --- Stage 3: temporal attention ---
  k_gemm_lr<<<1024, 128, 0, stream>>>(xsb, ta_w, leftb, rightb);
  k_attn<<<BB * WW, 128, 0, stream>>>(leftb, rightb, ta_b, ta_a, ta_bias, attnb);
  k_gemm_av<<<512, 128, 0, stream>>>(attnb, xsb, hgb);

  // --- Stage 4: GAT over top-k graph ---
  k_gemm_std<<<512, 128, 0, stream>>>(hgb, WW, WW, gat_w, hgatb);
  k_srcdst<<<1024, 128, 0, stream>>>(hgatb, gat_asrc, gat_adst, ssrcb, sdstb);
  k_fill_u32<<<cdiv(BnC, 256), 256, 0, stream>>>(emaxb, BnC, 0x007FFFFFu); // map(-inf)
  k_fill<<<cdiv(BnC, 256), 256, 0, stream>>>(denomb, BnC, 0.0f);
  k_edge_e<<<cdiv(nE_gat + BnC, 256), 256, 0, stream>>>(ei_gat, nE_gat, ssrcb, sdstb,
                                                        eedgeb, emaxb);
  k_edge_exp<<<cdiv(nE_gat + BnC, 256), 256, 0, stream>>>(ei_gat, nE_gat, eedgeb,
                                                          emaxb, denomb);
  k_fill<<<cdiv(BnC * WW, 256), 256, 0, stream>>>(goutb, BnC * WW, 0.0f);
  k_gat_agg<<<cdiv((nE_gat + BnC) * WW, 256), 256, 0, stream>>>(ei_gat, nE_gat, eedgeb,
                                                                denomb, hgatb, goutb);

  // --- Stage 5: final FC -> x_pre ---
  k_fc<<<1024, 128, 0, stream>>>(goutb, gat_b, fc_w, fc_b, outpre);
}